// OnlineLabelSmoothing_9414568313458
// MI455X (gfx1250) — compile-verified
//
#include <hip/hip_runtime.h>
#include <math.h>

// Online Label Smoothing — MI455X (gfx1250), wave32.
//
// Roofline: ~640 MB essential traffic / 23.3 TB/s ≈ 28 us floor; ~0.7 GFLOP
// total -> memory bound. Strategy: one wave per row; TDM stages the y_h row
// into LDS (async, TENSORcnt) while the wave streams y for the label argmax;
// log-softmax runs from LDS; the supervise-column dot runs on the matrix pipe
// as diag(accumulated v_wmma_f32_16x16x4_f32); update/idx_count are rare-event
// atomic scatters (P[pred==label] ~ 1/4096).

#define B_ROWS 16384
#define C_CLS  4096
#define ALPHA  0.5f
#define ROWS_PER_WG 8   // one row per wave, 8 waves (256 threads)

typedef __attribute__((ext_vector_type(2)))  float        v2f;
typedef __attribute__((ext_vector_type(8)))  float        v8f;
typedef __attribute__((ext_vector_type(4)))  unsigned int uv4;
typedef __attribute__((ext_vector_type(8)))  int          iv8;
typedef __attribute__((ext_vector_type(4)))  int          iv4;

__device__ __forceinline__ float waveReduceSum(float v) {
#pragma unroll
    for (int off = 16; off > 0; off >>= 1) v += __shfl_xor(v, off, 32);
    return v;
}

__device__ __forceinline__ void waveReduceArgmax(float& bm, int& bi) {
#pragma unroll
    for (int off = 16; off > 0; off >>= 1) {
        float ov = __shfl_xor(bm, off, 32);
        int   oi = __shfl_xor(bi, off, 32);
        if (ov > bm || (ov == bm && oi < bi)) { bm = ov; bi = oi; }  // first-occurrence tie-break
    }
}

__global__ void ols_zero_kernel(float* __restrict__ p, int n) {
    int i = blockIdx.x * blockDim.x + threadIdx.x;
    int stride = gridDim.x * blockDim.x;
    for (; i < n; i += stride) p[i] = 0.0f;
}

__global__ void __launch_bounds__(256)
ols_main_kernel(const float* __restrict__ y_h, const float* __restrict__ y,
                const float* __restrict__ sup, float* __restrict__ loss,
                float* __restrict__ update, float* __restrict__ idx_count) {
    extern __shared__ float lds[];  // ROWS_PER_WG * C_CLS floats (128 KB)

    const int lane = threadIdx.x & 31;
    const int wave = threadIdx.x >> 5;
    const int row  = blockIdx.x * ROWS_PER_WG + wave;
    float* myRow = lds + wave * C_CLS;

    // ---- 1) TDM: async DMA of this wave's y_h row (16 KB) into LDS ----
    {
        unsigned long long ga = (unsigned long long)(const void*)(y_h + (size_t)row * C_CLS);
        unsigned ldsAddr = (unsigned)(size_t)(void*)myRow;  // low 32 bits = LDS offset
        uv4 g0;
        g0[0] = 1u;                                          // count=1 (valid descriptor)
        g0[1] = ldsAddr;                                     // lds_addr [63:32]
        g0[2] = (unsigned)(ga & 0xFFFFFFFFu);                // global_addr lo
        g0[3] = (unsigned)((ga >> 32) & 0x01FFFFFFu) | 0x80000000u;  // addr hi | type=2
        iv8 g1;
        g1[0] = (int)(2u << 16);                             // data_size = 4 bytes
        g1[1] = (int)((C_CLS & 0xFFFF) << 16);               // tensor_dim0 low16
        g1[2] = (int)(((C_CLS >> 16) & 0xFFFF) | (1 << 16)); // dim0 hi16 | tensor_dim1=1 lo16
        g1[3] = (int)((C_CLS & 0xFFFF) << 16);               // dim1 hi16=0 | tile_dim0=C
        g1[4] = 1;                                           // tile_dim1=1, tile_dim2=0
        g1[5] = C_CLS;                                       // tensor_dim0_stride lo32
        g1[6] = 0;
        g1[7] = 0;
        iv4 g2 = {0, 0, 0, 0};
        iv4 g3 = {0, 0, 0, 0};
        iv8 g4 = {0, 0, 0, 0, 0, 0, 0, 0};                   // unused trailing group (2D tile)
        __builtin_amdgcn_tensor_load_to_lds(g0, g1, g2, g3, g4, 0);
    }

    // ---- 2) overlap: hard-label argmax over y row, streamed from global ----
    float ybm = -INFINITY; int ybi = 0x7FFFFFFF;
    {
        const float4* yrow = (const float4*)(y + (size_t)row * C_CLS);
#pragma unroll 4
        for (int i = lane; i < C_CLS / 4; i += 32) {
            float4 v = yrow[i];
            int b = i * 4;
            if (v.x > ybm) { ybm = v.x; ybi = b + 0; }
            if (v.y > ybm) { ybm = v.y; ybi = b + 1; }
            if (v.z > ybm) { ybm = v.z; ybi = b + 2; }
            if (v.w > ybm) { ybm = v.w; ybi = b + 3; }
        }
        waveReduceArgmax(ybm, ybi);
    }
    const int y_idx = ybi;

    // ---- 3) wait for this wave's DMA, then log-softmax from LDS ----
    __builtin_amdgcn_s_wait_tensorcnt(0);

    // sweep 1: row max + prediction argmax
    float hm = -INFINITY; int hp = 0x7FFFFFFF;
    for (int c = lane; c < C_CLS; c += 32) {
        float v = myRow[c];
        if (v > hm) { hm = v; hp = c; }
    }
    waveReduceArgmax(hm, hp);
    const float m = hm;
    const int pred = hp;

    // sweep 2: sum of exp(x - m)
    float se = 0.0f;
    for (int c = lane; c < C_CLS; c += 32) se += __expf(myRow[c] - m);
    const float Z = waveReduceSum(se);
    const float logZ = m + __logf(Z);  // log_p[c] = y_h[c] - logZ

    // ---- 4) smoothing dot on the matrix pipe:
    // dot = sum_c sup[c, y_idx] * y_h[c], via diag of accumulated 16x16x4 WMMA.
    // Lane mapping (f32 A 16x4): M = lane&15, K = {2*(lane>>4), 2*(lane>>4)+1}.
    // A[M,K] = supcol[64t + 4M + K];  B[K,N] = y_h[64t + 4N + K]  -> diag(D) sums chunk dots.
    v8f acc = {};
    float colsum = 0.0f;
    {
        const int mrow  = lane & 15;
        const int kbase = (lane >> 4) << 1;
        const int cofs  = mrow * 4 + kbase;
        for (int t = 0; t < C_CLS / 64; ++t) {
            int c0 = t * 64 + cofs;
            float a0 = sup[(size_t)c0 * C_CLS + y_idx];          // gathered supervise column
            float a1 = sup[(size_t)(c0 + 1) * C_CLS + y_idx];
            colsum += a0 + a1;                                   // each element loaded once/wave
            v2f A = {a0, a1};
            v2f Bv = {myRow[c0], myRow[c0 + 1]};                 // same index formula (N = lane&15)
            acc = __builtin_amdgcn_wmma_f32_16x16x4_f32(
                false, A, false, Bv, (short)0, acc, false, false);
        }
    }
    float d = 0.0f;
#pragma unroll
    for (int i = 0; i < 8; ++i) {  // diag lives at (lane i, vgpr i) and (lane i+24, vgpr i)
        float ci = acc[i];
        d += (lane == i || lane == i + 24) ? ci : 0.0f;
    }
    const float dot = waveReduceSum(d);
    const float supColSum = waveReduceSum(colsum);

    // ---- 5) losses ----
    if (lane == 0) {
        float hard_b = logZ - myRow[y_idx];                 // -log_p[y_idx]
        float soft_b = logZ * supColSum - dot;              // -sum sup[:,y_idx]*log_p
        float part = (ALPHA * hard_b + (1.0f - ALPHA) * soft_b) * (1.0f / (float)B_ROWS);
        atomicAdd(loss, part);
    }

    // ---- 6) rare-event buffer update (pred == label ~ 1/4096 of rows) ----
    if (pred == y_idx) {
        const float invZ = 1.0f / Z;
        for (int c = lane; c < C_CLS; c += 32) {
            float prob = __expf(myRow[c] - m) * invZ;
            atomicAdd(&update[(size_t)c * C_CLS + pred], prob);
        }
        if (lane == 0) atomicAdd(&idx_count[pred], 1.0f);
    }
}

extern "C" void kernel_launch(void* const* d_in, const int* in_sizes, int n_in,
                              void* d_out, int out_size, void* d_ws, size_t ws_size,
                              hipStream_t stream) {
    const float* y_h = (const float*)d_in[0];
    const float* y   = (const float*)d_in[1];
    const float* sup = (const float*)d_in[2];

    float* out       = (float*)d_out;
    float* loss      = out;                       // [1]
    float* update    = out + 1;                   // [C, C]
    float* idx_count = out + 1 + (size_t)C_CLS * C_CLS;  // [C]

    const int total = 1 + C_CLS * C_CLS + C_CLS;
    ols_zero_kernel<<<2048, 256, 0, stream>>>(out, total);

    const int grid = B_ROWS / ROWS_PER_WG;
    const size_t shmem = (size_t)ROWS_PER_WG * C_CLS * sizeof(float);  // 128 KB
    ols_main_kernel<<<grid, 256, shmem, stream>>>(y_h, y, sup, loss, update, idx_count);
}